// GCN_9783935500737
// MI455X (gfx1250) — compile-verified
//
#include <hip/hip_runtime.h>
#include <hip/hip_bf16.h>

#define D_FEAT 128

typedef __attribute__((ext_vector_type(2))) float v2f;
typedef __attribute__((ext_vector_type(8))) float v8f;

// ---------------------------------------------------------------------------
// Zero workspace (agg + deg). Grid-stride, fixed small grid.
// ---------------------------------------------------------------------------
__global__ void gcn_zero_ws(float* __restrict__ p, long long n) {
    long long i = (long long)blockIdx.x * blockDim.x + threadIdx.x;
    long long stride = (long long)gridDim.x * blockDim.x;
    for (; i < n; i += stride) p[i] = 0.0f;
}

// ---------------------------------------------------------------------------
// Edge scatter: agg[dst] += feature[src]; deg[dst] += 1.
// One wave per edge: 32 lanes x float4 = 128 features. b128 gather loads,
// global_atomic_add_f32 into agg (25.6MB, L2-resident on 192MB L2).
// ---------------------------------------------------------------------------
__global__ __launch_bounds__(256) void gcn_edge_scatter(
    const float* __restrict__ feature,
    const int*   __restrict__ src,
    const int*   __restrict__ dst,
    float*       __restrict__ agg,
    float*       __restrict__ deg,
    int n_edges)
{
    const int wave = threadIdx.x >> 5;            // 8 waves -> 8 edges / block
    const int lane = threadIdx.x & 31;
    const int e = blockIdx.x * 8 + wave;
    if (e >= n_edges) return;                     // uniform per wave

    const int s = src[e];
    const int d = dst[e];

    const float4 f = *(const float4*)(feature + (long long)s * D_FEAT + lane * 4);
    float* a = agg + (long long)d * D_FEAT + lane * 4;
    atomicAdd(a + 0, f.x);
    atomicAdd(a + 1, f.y);
    atomicAdd(a + 2, f.z);
    atomicAdd(a + 3, f.w);
    if (lane == 0) atomicAdd(&deg[d], 1.0f);
}

// ---------------------------------------------------------------------------
// Finalize (in place): h = deg>0 ? agg/deg : feature
// ---------------------------------------------------------------------------
__global__ __launch_bounds__(256) void gcn_finalize_mean(
    float*       __restrict__ agg,
    const float* __restrict__ deg,
    const float* __restrict__ feature,
    int n_nodes)
{
    long long i = (long long)blockIdx.x * blockDim.x + threadIdx.x;
    long long total = (long long)n_nodes * D_FEAT;
    if (i >= total) return;
    int node = (int)(i >> 7);                     // /128
    float dg = deg[node];
    agg[i] = (dg > 0.0f) ? (agg[i] / dg) : feature[i];
}

// ---------------------------------------------------------------------------
// out = relu(H @ W^T + b) via V_WMMA_F32_16X16X4_F32 (full f32 precision).
// Block = 256 threads = 8 waves. Wave w owns output tile
// rows [blockIdx*16, +16), cols [w*16, +16). K=128 -> 32 WMMA steps of K=4.
//
// f32 A 16x4 layout (ISA 7.12.2): lane m=lane&15; VGPR0/1 hold K = 2*(lane>>4)
// and +1. B 4x16 mirrors it: lane n=lane&15, same K split.
// C/D 16x16: VGPR j -> M = j + 8*(lane>>4), N = lane&15.
//
// Bias is folded into the accumulator init (every C element of lane L is
// column n0+(L&15)), so the epilogue is just relu + nontemporal store.
// The partial-tile guard is uniform per block -> branch-free fast path.
// ---------------------------------------------------------------------------
__global__ __launch_bounds__(256) void gcn_gemm_relu(
    const float* __restrict__ h,      // [n_rows, 128]
    const float* __restrict__ W,      // [128, 128] row-major (W[n][k])
    const float* __restrict__ bias,   // [128]
    float*       __restrict__ out,    // [n_rows, 128]
    int n_rows)
{
    const int wave = threadIdx.x >> 5;   // 0..7 -> output-column tile
    const int lane = threadIdx.x & 31;
    const int l15  = lane & 15;
    const int hi   = lane >> 4;          // 0 or 1 (K half of the 16x4 frag)

    const int  m0   = blockIdx.x * 16;
    const int  n0   = wave * 16;
    const bool full = (m0 + 16 <= n_rows);       // uniform per block

    int arow = m0 + l15;
    if (arow >= n_rows) arow = n_rows - 1;       // clamp, keep EXEC all-ones
    const int bcol = n0 + l15;

    const float* __restrict__ hrow = h + (long long)arow * D_FEAT + hi * 2;
    const float* __restrict__ wrow = W + (long long)bcol * D_FEAT + hi * 2;

    const float bv = bias[bcol];                 // N = lane&15 for both halves
    v8f c = { bv, bv, bv, bv, bv, bv, bv, bv };  // fold +b into accumulator

#pragma unroll 8
    for (int k = 0; k < D_FEAT; k += 4) {
        v2f a = *(const v2f*)(hrow + k);         // A[m, k+2*hi .. +1]
        v2f b = *(const v2f*)(wrow + k);         // B[k+2*hi .. +1, n] = W[n][k..]
        c = __builtin_amdgcn_wmma_f32_16x16x4_f32(
                /*neg_a=*/false, a, /*neg_b=*/false, b,
                /*c_mod=*/(short)0, c, /*reuse_a=*/false, /*reuse_b=*/false);
    }

    float* __restrict__ orow = out + (long long)(m0 + hi * 8) * D_FEAT + n0 + l15;
    if (full) {
#pragma unroll
        for (int j = 0; j < 8; ++j) {
            float v = c[j];
            __builtin_nontemporal_store(v > 0.0f ? v : 0.0f, orow + (long long)j * D_FEAT);
        }
    } else {
#pragma unroll
        for (int j = 0; j < 8; ++j) {
            if (m0 + hi * 8 + j < n_rows) {
                float v = c[j];
                orow[(long long)j * D_FEAT] = v > 0.0f ? v : 0.0f;
            }
        }
    }
}

// ---------------------------------------------------------------------------
// Launch
// inputs: 0=feature[N,128] f32, 1=W[128,128] f32, 2=b[128] f32,
//         3=src[E] i32, 4=dst[E] i32
// ---------------------------------------------------------------------------
extern "C" void kernel_launch(void* const* d_in, const int* in_sizes, int n_in,
                              void* d_out, int out_size, void* d_ws, size_t ws_size,
                              hipStream_t stream) {
    const float* feature = (const float*)d_in[0];
    const float* W       = (const float*)d_in[1];
    const float* b       = (const float*)d_in[2];
    const int*   src     = (const int*)d_in[3];
    const int*   dst     = (const int*)d_in[4];
    float*       out     = (float*)d_out;

    const int n_nodes = in_sizes[0] / D_FEAT;    // 50000
    const int n_edges = in_sizes[3];             // 800000

    float* agg = (float*)d_ws;                            // [n_nodes * 128]
    float* deg = agg + (long long)n_nodes * D_FEAT;       // [n_nodes]

    // 1) zero agg + deg
    const long long zn = (long long)n_nodes * D_FEAT + n_nodes;
    gcn_zero_ws<<<2048, 256, 0, stream>>>(agg, zn);

    // 2) scatter-add along edges (8 edges per 256-thread block)
    gcn_edge_scatter<<<(n_edges + 7) / 8, 256, 0, stream>>>(
        feature, src, dst, agg, deg, n_edges);

    // 3) mean + "keep own feature if no in-edges" (in place on agg)
    const long long total = (long long)n_nodes * D_FEAT;
    gcn_finalize_mean<<<(int)((total + 255) / 256), 256, 0, stream>>>(
        agg, deg, feature, n_nodes);

    // 4) relu(H @ W^T + b) with f32 WMMA
    gcn_gemm_relu<<<(n_nodes + 15) / 16, 256, 0, stream>>>(
        agg, W, b, out, n_nodes);
}